// DAS_BL_38268158607463
// MI455X (gfx1250) — compile-verified
//
#include <hip/hip_runtime.h>
#include <hip/hip_bf16.h>
#include <math.h>
#include <float.h>

typedef float v2f __attribute__((ext_vector_type(2)));
typedef float v8f __attribute__((ext_vector_type(8)));

#define B_ROWS 8192
#define D_DIM  256
#define C_CLS  5994
#define H_ROWS 4096
#define N_COL_TILES ((C_CLS + 15) / 16)   // 375
#define HEAD_SCALE 30.0f
#define MARGIN 2.0f

__device__ __forceinline__ float waveReduceSum(float v) {
#pragma unroll
  for (int d = 16; d > 0; d >>= 1) v += __shfl_xor(v, d, 32);
  return v;
}

// ---------------------------------------------------------------- norms ----
__global__ void __launch_bounds__(256)
norms_kernel(const float* __restrict__ emb, const float* __restrict__ W,
             float* __restrict__ inv_e, float* __restrict__ inv_w) {
  const int lane = threadIdx.x & 31;
  const int wave = threadIdx.x >> 5;
  const int i = blockIdx.x * 8 + wave;
  const int R = B_ROWS + C_CLS;
  if (i >= R) return;
  const float* row = (i < B_ROWS) ? (emb + (size_t)i * D_DIM)
                                  : (W + (size_t)(i - B_ROWS) * D_DIM);
  float ssum = 0.f;
#pragma unroll
  for (int j = 0; j < 8; ++j) { float v = row[lane + 32 * j]; ssum = fmaf(v, v, ssum); }
  ssum = waveReduceSum(ssum);
  if (lane == 0) {
    float inv = 1.f / (sqrtf(ssum) + 1e-12f);
    if (i < B_ROWS) inv_e[i] = inv; else inv_w[i - B_ROWS] = inv;
  }
}

// ------------------------------------------- fused GEMM + online softmax ----
// Block: 256 threads (8 waves), owns 16 rows of emb. Waves stride over
// 16-column tiles of W. Per tile: 64 x V_WMMA_F32_16X16X4_F32, then a
// BRANCHLESS online (max, sumexp, argmax) update per lane-component.
__global__ void __launch_bounds__(256)
gemm_softmax_kernel(const float* __restrict__ emb, const float* __restrict__ W,
                    const float* __restrict__ inv_e, const float* __restrict__ inv_w,
                    float* __restrict__ row_m, float* __restrict__ row_s,
                    int* __restrict__ row_arg) {
  __shared__ float sA[16 * 260];     // 16 rows x 256, padded stride
  __shared__ float sm[8][16];
  __shared__ float ss[8][16];
  __shared__ int   sbi[8][16];

  const int tid  = threadIdx.x;
  const int lane = tid & 31;
  const int wave = tid >> 5;
  const int half = lane >> 4;        // 0: rows 0..7 / K 0,1   1: rows 8..15 / K 2,3
  const int l16  = lane & 15;
  const int rowBase = blockIdx.x * 16;

#pragma unroll
  for (int t = 0; t < 16; ++t)
    sA[t * 260 + tid] = emb[(size_t)(rowBase + t) * D_DIM + tid];
  __syncthreads();

  float ie[8];
#pragma unroll
  for (int r = 0; r < 8; ++r) ie[r] = inv_e[rowBase + half * 8 + r] * HEAD_SCALE;

  float m[8], s[8]; int bi[8];
#pragma unroll
  for (int r = 0; r < 8; ++r) { m[r] = -FLT_MAX; s[r] = 0.f; bi[r] = 0x7fffffff; }

  const float* ap = &sA[l16 * 260 + half * 2];   // A fragment: lane row l16, K pair by half

  for (int ct = wave; ct < N_COL_TILES; ct += 8) {
    const int col  = ct * 16 + l16;
    const int colc = (col < C_CLS) ? col : (C_CLS - 1);
    const float* wrow = W + (size_t)colc * D_DIM;
    const float iw = inv_w[colc];

    v8f c = {};
#pragma unroll 8
    for (int k = 0; k < D_DIM; k += 4) {
      v2f a, b;
      a.x = ap[k];               a.y = ap[k + 1];
      b.x = wrow[k + half * 2];  b.y = wrow[k + half * 2 + 1];
      c = __builtin_amdgcn_wmma_f32_16x16x4_f32(false, a, false, b, (short)0, c,
                                                false, false);
    }

    const bool valid = (col < C_CLS);
#pragma unroll
    for (int r = 0; r < 8; ++r) {
      float v = valid ? c[r] * iw * ie[r] : -INFINITY;
      float d = v - m[r];
      float e = __expf(-fabsf(d));     // exp(min(d,-d)) <= 1, never NaN here
      bool gt = d > 0.f;
      s[r]  = gt ? fmaf(s[r], e, 1.f) : (s[r] + e);
      bi[r] = gt ? col : bi[r];
      m[r]  = fmaxf(m[r], v);
    }
  }

  // reduce (m,s,argmax) across the 16 lanes holding one row
#pragma unroll
  for (int d = 1; d < 16; d <<= 1) {
#pragma unroll
    for (int r = 0; r < 8; ++r) {
      float m2 = __shfl_xor(m[r], d, 32);
      float s2 = __shfl_xor(s[r], d, 32);
      int   i2 = __shfl_xor(bi[r], d, 32);
      float nm = fmaxf(m[r], m2);
      float ns = s[r] * __expf(m[r] - nm) + s2 * __expf(m2 - nm);
      bi[r] = (m2 > m[r] || (m2 == m[r] && i2 < bi[r])) ? i2 : bi[r];
      m[r] = nm; s[r] = ns;
    }
  }
  if (l16 == 0) {
#pragma unroll
    for (int r = 0; r < 8; ++r) {
      sm[wave][half * 8 + r]  = m[r];
      ss[wave][half * 8 + r]  = s[r];
      sbi[wave][half * 8 + r] = bi[r];
    }
  }
  __syncthreads();

  if (tid < 16) {   // merge the 8 waves' column subsets
    float fm = sm[0][tid], fs = ss[0][tid]; int fi = sbi[0][tid];
    for (int w = 1; w < 8; ++w) {
      float m2 = sm[w][tid], s2 = ss[w][tid]; int i2 = sbi[w][tid];
      float nm = fmaxf(fm, m2);
      fs = fs * __expf(fm - nm) + s2 * __expf(m2 - nm);
      fi = (m2 > fm || (m2 == fm && i2 < fi)) ? i2 : fi;
      fm = nm;
    }
    row_m[rowBase + tid]   = fm;
    row_s[rowBase + tid]   = fs;
    row_arg[rowBase + tid] = fi;
  }
}

// ------------------------------------------------ target logit + CE/acc ----
__global__ void __launch_bounds__(256)
finalize_kernel(const float* __restrict__ emb, const float* __restrict__ W,
                const int* __restrict__ y,
                const float* __restrict__ inv_e, const float* __restrict__ inv_w,
                const float* __restrict__ row_m, const float* __restrict__ row_s,
                const int* __restrict__ row_arg, float* __restrict__ accum) {
  const int lane = threadIdx.x & 31;
  const int wave = threadIdx.x >> 5;
  const int i = blockIdx.x * 8 + wave;
  if (i >= B_ROWS) return;
  const int yi = y[i];
  const float* e = emb + (size_t)i * D_DIM;
  const float* w = W + (size_t)yi * D_DIM;
  float dot = 0.f;
#pragma unroll
  for (int j = 0; j < 8; ++j) { int k = lane + 32 * j; dot = fmaf(e[k], w[k], dot); }
  dot = waveReduceSum(dot);
  if (lane == 0) {
    float logit_y = HEAD_SCALE * inv_e[i] * inv_w[yi] * dot;
    atomicAdd(&accum[0], row_m[i] + __logf(row_s[i]) - logit_y);
    atomicAdd(&accum[2], (row_arg[i] == yi) ? 1.f : 0.f);
  }
}

// --------------------------------------------------------------- DAS ----
__global__ void __launch_bounds__(256)
das_kernel(const float* __restrict__ emb, const int* __restrict__ y_d,
           float* __restrict__ accum) {
  const int lane = threadIdx.x & 31;
  const int wave = threadIdx.x >> 5;
  const int i = blockIdx.x * 8 + wave;
  if (i >= H_ROWS) return;
  const int j = (i + 1) & (H_ROWS - 1);
  const float mi = (y_d[i] == 0) ? 1.f : 0.f;
  const float mj = (y_d[j] == 0) ? 1.f : 0.f;
  const bool tHi = (y_d[i + H_ROWS] != 0);   // scatter: row i+H wins target slot i
  const bool tLo = (y_d[i] != 0);
  const float* ei = emb + (size_t)i * D_DIM;
  const float* ej = emb + (size_t)j * D_DIM;
  const float* et = emb + (size_t)(i + H_ROWS) * D_DIM;
  float sp = 0.f, sn = 0.f;
#pragma unroll
  for (int q = 0; q < 8; ++q) {
    int k = lane + 32 * q;
    float a = mi * ei[k];
    float b = mj * ej[k];
    float t = tHi ? et[k] : (tLo ? ei[k] : 0.f);
    float dp = a - b, dn = a - t;
    sp = fmaf(dp, dp, sp);
    sn = fmaf(dn, dn, sn);
  }
  sp = waveReduceSum(sp);
  sn = waveReduceSum(sn);
  if (lane == 0) {
    float d_pos = sqrtf(sp), d_neg = sqrtf(sn);
    float hp = fmaxf(MARGIN - d_pos, 0.f);
    atomicAdd(&accum[1], hp * hp + sn);          // d_neg^2 == sn
    atomicAdd(&accum[3], d_pos + d_neg);
  }
}

// ------------------------------------------------------------- glue ----
__global__ void init_kernel(float* accum) {
  if (threadIdx.x < 4) accum[threadIdx.x] = 0.f;
}
__global__ void out_kernel(const float* __restrict__ accum, float* __restrict__ out) {
  if (threadIdx.x == 0) {
    const float invB = 1.f / (float)B_ROWS;
    out[0] = accum[0] * invB;   // loss_c
    out[1] = accum[1] * invB;   // das_loss
    out[2] = accum[2] * invB;   // acc
    out[3] = accum[3] * invB;   // das_dist_mean
  }
}

extern "C" void kernel_launch(void* const* d_in, const int* in_sizes, int n_in,
                              void* d_out, int out_size, void* d_ws, size_t ws_size,
                              hipStream_t stream) {
  const float* emb = (const float*)d_in[0];
  const float* W   = (const float*)d_in[1];
  const int*   y   = (const int*)d_in[2];
  const int*   y_d = (const int*)d_in[3];
  float* out = (float*)d_out;

  float* inv_e   = (float*)d_ws;            // 8192
  float* inv_w   = inv_e + B_ROWS;          // 6016 (padded)
  float* row_m   = inv_w + 6016;            // 8192
  float* row_s   = row_m + B_ROWS;          // 8192
  int*   row_arg = (int*)(row_s + B_ROWS);  // 8192
  float* accum   = (float*)(row_arg + B_ROWS); // 4

  hipLaunchKernelGGL(init_kernel, dim3(1), dim3(32), 0, stream, accum);
  const int Rn = (B_ROWS + C_CLS + 7) / 8;
  hipLaunchKernelGGL(norms_kernel, dim3(Rn), dim3(256), 0, stream, emb, W, inv_e, inv_w);
  hipLaunchKernelGGL(gemm_softmax_kernel, dim3(B_ROWS / 16), dim3(256), 0, stream,
                     emb, W, inv_e, inv_w, row_m, row_s, row_arg);
  hipLaunchKernelGGL(finalize_kernel, dim3(B_ROWS / 8), dim3(256), 0, stream,
                     emb, W, y, inv_e, inv_w, row_m, row_s, row_arg, accum);
  hipLaunchKernelGGL(das_kernel, dim3(H_ROWS / 8), dim3(256), 0, stream, emb, y_d, accum);
  hipLaunchKernelGGL(out_kernel, dim3(1), dim3(32), 0, stream, accum, out);
}